// GlobalModel_19404662243987
// MI455X (gfx1250) — compile-verified
//
#include <hip/hip_runtime.h>
#include <hip/hip_bf16.h>

typedef __attribute__((ext_vector_type(2))) float v2f;
typedef __attribute__((ext_vector_type(8))) float v8f;

#define N_GRAPHS    4096
#define NODE_SIZE   512
#define GLOBAL_SIZE 256
#define HIDDEN_SIZE 1024
#define INPUT_SIZE  768   // GLOBAL_SIZE + NODE_SIZE

// workspace layout (in floats)
// All WMMA operands are stored "pair-packed": P[k2][row][p] holds element [row][2*k2+p],
// so one lane's A/B fragment half (two consecutive K values) is a single 8-byte load.
#define WS_SUMS   0
#define WS_COUNTS (N_GRAPHS * NODE_SIZE)                  // 2,097,152
#define WS_AP     (WS_COUNTS + N_GRAPHS)                  // Ap : INPUT_SIZE  x N_GRAPHS (pair-packed)
#define WS_HTP    (WS_AP  + INPUT_SIZE  * N_GRAPHS)       // Htp: HIDDEN_SIZE x N_GRAPHS (pair-packed)
#define WS_W1P    (WS_HTP + HIDDEN_SIZE * N_GRAPHS)       // W1p: INPUT_SIZE  x HIDDEN_SIZE (pair-packed)
#define WS_W2P    (WS_W1P + INPUT_SIZE  * HIDDEN_SIZE)    // W2p: HIDDEN_SIZE x GLOBAL_SIZE (pair-packed)
// total = WS_W2P + HIDDEN_SIZE*GLOBAL_SIZE = 10,489,856 floats ~= 42 MB

// branchless SELU: for v>0 the exp term is exactly 0; for v<=0 the max term is 0.
__device__ __forceinline__ float selu_f(float v) {
    const float SC = 1.0507009873554805f;
    const float AL = 1.6732632423543772f;
    return SC * (fmaxf(v, 0.f) + AL * (__expf(fminf(v, 0.f)) - 1.f));
}

// ---------------------------------------------------------------- zero scratch
__global__ void zero_kernel(float* __restrict__ p, int n) {
    int i = blockIdx.x * blockDim.x + threadIdx.x;
    if (i < n) p[i] = 0.0f;
}

// ---------------------------------------------------------------- phase 1: segment sums
// batch is sorted: accumulate in registers, flush with atomics only at segment boundaries.
// blockDim = 128; thread t owns features [4t, 4t+3] (float4).
__global__ void segsum_kernel(const float* __restrict__ x, const int* __restrict__ batch,
                              float* __restrict__ sums, float* __restrict__ counts,
                              int n_nodes, int nodes_per_block) {
    const int t = threadIdx.x;                 // 0..127
    int start = blockIdx.x * nodes_per_block;
    int end   = start + nodes_per_block;
    if (end > n_nodes) end = n_nodes;
    if (start >= end) return;

    float4 acc = make_float4(0.f, 0.f, 0.f, 0.f);
    float  cnt = 0.f;
    int    cur = -1;

    for (int n = start; n < end; ++n) {
        int g = batch[n];
        if (g != cur) {
            if (cur >= 0) {
                float* s = sums + (size_t)cur * NODE_SIZE + 4 * t;
                atomicAdd(s + 0, acc.x); atomicAdd(s + 1, acc.y);
                atomicAdd(s + 2, acc.z); atomicAdd(s + 3, acc.w);
                if (t == 0) atomicAdd(counts + cur, cnt);
            }
            acc = make_float4(0.f, 0.f, 0.f, 0.f);
            cnt = 0.f;
            cur = g;
        }
        float4 v = ((const float4*)(x + (size_t)n * NODE_SIZE))[t];
        acc.x += v.x; acc.y += v.y; acc.z += v.z; acc.w += v.w;
        cnt += 1.f;
    }
    if (cur >= 0) {
        float* s = sums + (size_t)cur * NODE_SIZE + 4 * t;
        atomicAdd(s + 0, acc.x); atomicAdd(s + 1, acc.y);
        atomicAdd(s + 2, acc.z); atomicAdd(s + 3, acc.w);
        if (t == 0) atomicAdd(counts + cur, cnt);
    }
}

// ---------------------------------------------------------------- build pair-packed concat A
// Ap[(k2*N_GRAPHS + row)*2 + p] = concat(u, mean)[row][2*k2+p]
__global__ void concat_pack_kernel(const float* __restrict__ u, const float* __restrict__ sums,
                                   const float* __restrict__ counts, float* __restrict__ Ap) {
    int i = blockIdx.x * blockDim.x + threadIdx.x;   // index directly into Ap (coalesced writes)
    if (i >= INPUT_SIZE * N_GRAPHS) return;
    int p   = i & 1;
    int rk  = i >> 1;
    int row = rk & (N_GRAPHS - 1);
    int k   = ((rk >> 12) << 1) + p;                 // 2*k2 + p   (N_GRAPHS == 2^12)
    float v;
    if (k < GLOBAL_SIZE) {
        v = u[(size_t)row * GLOBAL_SIZE + k];
    } else {
        float inv = 1.0f / fmaxf(counts[row], 1.0f);
        v = sums[(size_t)row * NODE_SIZE + (k - GLOBAL_SIZE)] * inv;
    }
    Ap[i] = v;
}

// ---------------------------------------------------------------- pair-pack a row-major weight [K][N]
__global__ void pack_w_kernel(const float* __restrict__ W, float* __restrict__ Wp, int K, int N) {
    int i = blockIdx.x * blockDim.x + threadIdx.x;
    if (i >= K * N) return;
    int p   = i & 1;
    int ck  = i >> 1;
    int col = ck % N;
    int k   = ((ck / N) << 1) + p;
    Wp[i] = W[(size_t)k * N + col];
}

// ---------------------------------------------------------------- GEMM1: h = selu(A @ W1 + b1)
// One wave computes a 16x64 output tile (4 N-tiles): A fragment reused 4x per K-step.
// Inner loop: 1 b64 A-load + 4 b64 B-loads + 4 WMMAs.
__global__ void gemm1_selu_kernel(const float* __restrict__ Ap, const float* __restrict__ W1p,
                                  const float* __restrict__ b1, float* __restrict__ Htp) {
    const int lane  = threadIdx.x & 31;
    const int wave  = blockIdx.x * (blockDim.x >> 5) + (threadIdx.x >> 5);
    const int NGRP  = HIDDEN_SIZE / 64;              // 16 groups of 4 tiles
    const int tileM = wave / NGRP;                   // 0..255
    const int ng    = wave % NGRP;                   // 0..15
    const int l15   = lane & 15;
    const int kh    = lane >> 4;                     // pair index: 0 (lanes 0-15), 1 (lanes 16-31)
    const int row   = tileM * 16 + l15;

    int col[4];
    v8f c[4];
#pragma unroll
    for (int j = 0; j < 4; ++j) {
        col[j] = (ng * 4 + j) * 16 + l15;
        float bias = b1[col[j]];
#pragma unroll
        for (int r = 0; r < 8; ++r) c[j][r] = bias;
    }

    const v2f* A2 = (const v2f*)Ap;
    const v2f* B2 = (const v2f*)W1p;
    for (int k0 = 0; k0 < INPUT_SIZE; k0 += 4) {
        int k2 = (k0 >> 1) + kh;
        v2f a = A2[(size_t)k2 * N_GRAPHS + row];
#pragma unroll
        for (int j = 0; j < 4; ++j) {
            v2f b = B2[(size_t)k2 * HIDDEN_SIZE + col[j]];
            c[j] = __builtin_amdgcn_wmma_f32_16x16x4_f32(false, a, false, b, (short)0, c[j],
                                                         false, false);
        }
    }

#pragma unroll
    for (int j = 0; j < 4; ++j) {
#pragma unroll
        for (int r = 0; r < 8; ++r) {
            int rowOut = tileM * 16 + r + (kh << 3); // VGPR r: M=r (lanes 0-15), M=r+8 (16-31)
            float v = selu_f(c[j][r]);
            // pair-packed transposed store: Htp[(k2h*N_GRAPHS + row)*2 + p] with k=col[j]
            Htp[((size_t)(col[j] >> 1) * N_GRAPHS + rowOut) * 2 + (col[j] & 1)] = v;
        }
    }
}

// ---------------------------------------------------------------- GEMM2: out = h @ W2 + b2
__global__ void gemm2_kernel(const float* __restrict__ Htp, const float* __restrict__ W2p,
                             const float* __restrict__ b2, float* __restrict__ out) {
    const int lane  = threadIdx.x & 31;
    const int wave  = blockIdx.x * (blockDim.x >> 5) + (threadIdx.x >> 5);
    const int NGRP  = GLOBAL_SIZE / 64;              // 4 groups of 4 tiles
    const int tileM = wave / NGRP;                   // 0..255
    const int ng    = wave % NGRP;                   // 0..3
    const int l15   = lane & 15;
    const int kh    = lane >> 4;
    const int row   = tileM * 16 + l15;

    int col[4];
    v8f c[4];
#pragma unroll
    for (int j = 0; j < 4; ++j) {
        col[j] = (ng * 4 + j) * 16 + l15;
        float bias = b2[col[j]];
#pragma unroll
        for (int r = 0; r < 8; ++r) c[j][r] = bias;
    }

    const v2f* A2 = (const v2f*)Htp;
    const v2f* B2 = (const v2f*)W2p;
    for (int k0 = 0; k0 < HIDDEN_SIZE; k0 += 4) {
        int k2 = (k0 >> 1) + kh;
        v2f a = A2[(size_t)k2 * N_GRAPHS + row];
#pragma unroll
        for (int j = 0; j < 4; ++j) {
            v2f b = B2[(size_t)k2 * GLOBAL_SIZE + col[j]];
            c[j] = __builtin_amdgcn_wmma_f32_16x16x4_f32(false, a, false, b, (short)0, c[j],
                                                         false, false);
        }
    }

#pragma unroll
    for (int j = 0; j < 4; ++j) {
#pragma unroll
        for (int r = 0; r < 8; ++r) {
            int rowOut = tileM * 16 + r + (kh << 3);
            out[(size_t)rowOut * GLOBAL_SIZE + col[j]] = c[j][r];
        }
    }
}

// ---------------------------------------------------------------- launch
extern "C" void kernel_launch(void* const* d_in, const int* in_sizes, int n_in,
                              void* d_out, int out_size, void* d_ws, size_t ws_size,
                              hipStream_t stream) {
    const float* x     = (const float*)d_in[0];
    // d_in[1] edge_index, d_in[2] edge_attr: unused by the reference computation
    const float* u     = (const float*)d_in[3];
    const int*   batch = (const int*)  d_in[4];
    const float* W1    = (const float*)d_in[5];
    const float* b1    = (const float*)d_in[6];
    const float* W2    = (const float*)d_in[7];
    const float* b2    = (const float*)d_in[8];
    float*       out   = (float*)d_out;

    const int n_nodes = in_sizes[4];

    float* ws     = (float*)d_ws;
    float* sums   = ws + WS_SUMS;
    float* counts = ws + WS_COUNTS;
    float* Ap     = ws + WS_AP;
    float* Htp    = ws + WS_HTP;
    float* W1p    = ws + WS_W1P;
    float* W2p    = ws + WS_W2P;

    // 1) zero sums + counts (must re-zero every call: we accumulate into them)
    {
        int n = WS_COUNTS + N_GRAPHS;
        zero_kernel<<<(n + 255) / 256, 256, 0, stream>>>(ws, n);
    }

    // 2) pair-pack weights (independent of phase 1; cheap: 4 MB total)
    {
        int n1 = INPUT_SIZE * HIDDEN_SIZE;
        pack_w_kernel<<<(n1 + 255) / 256, 256, 0, stream>>>(W1, W1p, INPUT_SIZE, HIDDEN_SIZE);
        int n2 = HIDDEN_SIZE * GLOBAL_SIZE;
        pack_w_kernel<<<(n2 + 255) / 256, 256, 0, stream>>>(W2, W2p, HIDDEN_SIZE, GLOBAL_SIZE);
    }

    // 3) segment sums over sorted batch (register accumulation, boundary atomics)
    {
        const int blocks = 4096;
        int npb = (n_nodes + blocks - 1) / blocks;
        segsum_kernel<<<blocks, 128, 0, stream>>>(x, batch, sums, counts, n_nodes, npb);
    }

    // 4) concat [u | mean] into pair-packed A
    {
        int n = INPUT_SIZE * N_GRAPHS;
        concat_pack_kernel<<<(n + 255) / 256, 256, 0, stream>>>(u, sums, counts, Ap);
    }

    // 5) h = selu(A @ W1 + b1), stored pair-packed/transposed (4096 waves, 8 waves/block)
    {
        int waves  = (N_GRAPHS / 16) * (HIDDEN_SIZE / 64);   // 4096
        gemm1_selu_kernel<<<waves / 8, 256, 0, stream>>>(Ap, W1p, b1, Htp);
    }

    // 6) out = h @ W2 + b2  (1024 waves)
    {
        int waves  = (N_GRAPHS / 16) * (GLOBAL_SIZE / 64);   // 1024
        gemm2_kernel<<<waves / 8, 256, 0, stream>>>(Htp, W2p, b2, out);
    }
}